// RegressionHead_22127671509438
// MI455X (gfx1250) — compile-verified
//
#include <hip/hip_runtime.h>
#include <math.h>

#define W   320
#define WP  321
#define NT  256

typedef __attribute__((ext_vector_type(2))) float v2f;
typedef __attribute__((ext_vector_type(8))) float v8f;

// log(2w) = log(640), log_mu = log(1/(2w)), log_mu_dustbin = log(1/2)
#define LOG2W    6.46146816f
#define LOG_MU  (-6.46146816f)
#define LOG_HALF (-0.69314718f)

__device__ __forceinline__ float wave_max32(float m) {
#pragma unroll
  for (int off = 16; off; off >>= 1) m = fmaxf(m, __shfl_xor(m, off, 32));
  return m;
}

// block-wide logsumexp over arr[0..n) using shared scratch red[NT]
__device__ float block_lse(const float* arr, int n, float* red) {
  const int t = threadIdx.x;
  float m = -INFINITY;
  for (int k = t; k < n; k += NT) m = fmaxf(m, arr[k]);
  red[t] = m;
  __syncthreads();
  for (int s = NT / 2; s > 0; s >>= 1) {
    if (t < s) red[t] = fmaxf(red[t], red[t + s]);
    __syncthreads();
  }
  m = red[0];
  __syncthreads();
  float ssum = 0.f;
  for (int k = t; k < n; k += NT) ssum += __expf(arr[k] - m);
  red[t] = ssum;
  __syncthreads();
  for (int s = NT / 2; s > 0; s >>= 1) {
    if (t < s) red[t] += red[t + s];
    __syncthreads();
  }
  float r = m + __logf(red[0]);
  __syncthreads();
  return r;
}

__global__ __launch_bounds__(NT) void sinkhorn_ot_kernel(
    const float* __restrict__ attn, const unsigned char* __restrict__ occm,
    const float* __restrict__ phi_p, float* __restrict__ out) {
  __shared__ float u[WP + 3];
  __shared__ float v[WP + 3];
  __shared__ float mrow[W];
  __shared__ float red[NT];

  const int t = threadIdx.x;
  const int lane = t & 31;
  const int wave = t >> 5;
  const int bh = blockIdx.x;  // 0..359  (bs*h flattened)
  const float* A = attn + (size_t)bh * (size_t)(W * W);
  const float phi = phi_p[0];

  for (int k = t; k < WP; k += NT) { u[k] = 0.f; v[k] = 0.f; }
  __syncthreads();

  v2f ones; ones.x = 1.0f; ones.y = 1.0f;

  for (int iter = 0; iter < 10; ++iter) {
    // ---------------- v update: v[j] = log_nu[j] - lse_i(sim[i][j] + u[i])
    float lse_u = block_lse(u, WP, red);
    float u320 = u[W];
#pragma unroll
    for (int pass = 0; pass < 2; ++pass) {
      int j = (pass == 0) ? t : t + NT;
      if (j < W) {
        float mc = phi + u320;  // dustbin row i=320 contributes phi+u[320]
#pragma unroll 4
        for (int i = 0; i < W; ++i) mc = fmaxf(mc, A[(size_t)i * W + j] + u[i]);
        float s = __expf(phi + u320 - mc);
#pragma unroll 4
        for (int i = 0; i < W; ++i) s += __expf(A[(size_t)i * W + j] + u[i] - mc);
        v[j] = LOG_MU - (mc + __logf(s));
      }
    }
    if (t == 0) v[W] = LOG_HALF - (phi + lse_u);
    __syncthreads();

    // ---------------- u update: u[i] = log_mu[i] - lse_j(sim[i][j] + v[j])
    float lse_v = block_lse(v, WP, red);
    float v320 = v[W];
    // row maxes (wave per row, coalesced, shuffle reduce)
    for (int r = wave; r < W; r += 8) {
      const float* Ar = A + (size_t)r * W;
      float m = phi + v320;  // dustbin column j=320
      for (int j0 = lane; j0 < W; j0 += 32) m = fmaxf(m, Ar[j0] + v[j0]);
      m = wave_max32(m);
      if (lane == 0) mrow[r] = m;
    }
    __syncthreads();

    // WMMA row-sum pass: D = A_tile x ones + C accumulates 16 row-sums.
    // A-operand layout (ISA 7.12.2, 32-bit A 16x4): lane L holds row (L&15),
    // K = (L>>4)*2 in v0 and K+1 in v1. B = all-ones is layout invariant.
    for (int tile = wave; tile < W / 16; tile += 8) {
      const int i0 = tile * 16;
      const int row = i0 + (lane & 15);
      const int kb = (lane >> 4) * 2;
      const float rm = mrow[row];
      const float* Ar = A + (size_t)row * W;
      v8f acc = {0.f, 0.f, 0.f, 0.f, 0.f, 0.f, 0.f, 0.f};
      for (int j = 0; j < W; j += 4) {
        v2f a;
        a.x = __expf(Ar[j + kb]     + v[j + kb]     - rm);
        a.y = __expf(Ar[j + kb + 1] + v[j + kb + 1] - rm);
        acc = __builtin_amdgcn_wmma_f32_16x16x4_f32(
            false, a, false, ones, (short)0, acc, false, false);
      }
      // D layout: lane 0 -> rows i0+0..7 in acc[0..7]; lane 16 -> rows i0+8..15
      if (lane == 0 || lane == 16) {
        const int base = i0 + ((lane >> 4) << 3);
#pragma unroll
        for (int r = 0; r < 8; ++r) {
          const int rr = base + r;
          float total = acc[r] + __expf(phi + v320 - mrow[rr]);  // dustbin col
          u[rr] = LOG_MU - (mrow[rr] + __logf(total));
        }
      }
    }
    if (t == 0) u[W] = LOG_HALF - (phi + lse_v);
    __syncthreads();
  }

  // ---------------- outputs ----------------
  const float u320 = u[W];
  const float v320 = v[W];
  float* OT = out + 2 * 115200 + (size_t)bh * (size_t)(WP * WP);

  for (int r = wave; r < WP; r += 8) {
    if (r < W) {
      const float ui = u[r];
      const float* Ar = A + (size_t)r * W;
      float* Or = OT + (size_t)r * WP;
      float bestv = -INFINITY;
      int bestj = 0;
      for (int j0 = lane; j0 < W; j0 += 32) {
        float val = __expf(Ar[j0] + ui + v[j0] + LOG2W);
        Or[j0] = val;
        if (val > bestv) { bestv = val; bestj = j0; }
      }
      // wave argmax, first-index on ties (matches jnp.argmax)
#pragma unroll
      for (int off = 16; off; off >>= 1) {
        float ov = __shfl_xor(bestv, off, 32);
        int oj = __shfl_xor(bestj, off, 32);
        if (ov > bestv || (ov == bestv && oj < bestj)) { bestv = ov; bestj = oj; }
      }
      if (lane == 0) {
        Or[W] = __expf(phi + ui + v320 + LOG2W);  // dustbin column
        const int high = bestj;
        float wsum = 0.f, dsum = 0.f;
#pragma unroll
        for (int d = -1; d <= 1; ++d) {
          const int c = high + d;
          if (c >= 0 && c < W) {
            float wc = __expf(Ar[c] + ui + v[c] + LOG2W);
            float pc = fmaxf((float)(r - c), 0.f);  // pos_shift = max(i-j,0)
            wsum += wc;
            dsum += wc * pc;
          }
        }
        const bool om = occm[bh * W + r] != 0;
        const float norm = om ? 1.f : wsum;
        out[bh * W + r] = dsum / norm;             // disp_pred_low_res
        out[115200 + bh * W + r] = 1.f - norm;     // occ_pred_low_res
      }
    } else {
      // dustbin row i = 320
      float* Or = OT + (size_t)W * WP;
      for (int j0 = lane; j0 < W; j0 += 32)
        Or[j0] = __expf(phi + u320 + v[j0] + LOG2W);
      if (lane == 0) Or[W] = __expf(phi + u320 + v320 + LOG2W);
    }
  }
}

extern "C" void kernel_launch(void* const* d_in, const int* in_sizes, int n_in,
                              void* d_out, int out_size, void* d_ws, size_t ws_size,
                              hipStream_t stream) {
  (void)in_sizes; (void)n_in; (void)d_ws; (void)ws_size; (void)out_size;
  const float* attn = (const float*)d_in[0];
  const unsigned char* occm = (const unsigned char*)d_in[1];  // jnp.bool_ = 1 byte
  const float* phi = (const float*)d_in[2];
  float* out = (float*)d_out;
  // 360 blocks = bs*h, one 321x321 Sinkhorn problem per workgroup
  hipLaunchKernelGGL(sinkhorn_ot_kernel, dim3(360), dim3(NT), 0, stream,
                     attn, occm, phi, out);
}